// HGCN_15522011808429
// MI455X (gfx1250) — compile-verified
//
#include <hip/hip_runtime.h>

#define EPSV 1e-15f
#define MAXN 0.996f   /* (1 - 4e-3) / sqrt(c=1) */

typedef __attribute__((ext_vector_type(2))) float v2f;
typedef __attribute__((ext_vector_type(8))) float v8f;

// wave32 all-lanes sum via DS_SWIZZLE_B32 group-of-32 XOR patterns:
// offset = (xor_mask << 10) | and_mask(0x1F)
__device__ __forceinline__ float wsum(float v) {
  v += __builtin_bit_cast(float, __builtin_amdgcn_ds_swizzle(__builtin_bit_cast(int, v), 0x401F)); // xor 16
  v += __builtin_bit_cast(float, __builtin_amdgcn_ds_swizzle(__builtin_bit_cast(int, v), 0x201F)); // xor 8
  v += __builtin_bit_cast(float, __builtin_amdgcn_ds_swizzle(__builtin_bit_cast(int, v), 0x101F)); // xor 4
  v += __builtin_bit_cast(float, __builtin_amdgcn_ds_swizzle(__builtin_bit_cast(int, v), 0x081F)); // xor 2
  v += __builtin_bit_cast(float, __builtin_amdgcn_ds_swizzle(__builtin_bit_cast(int, v), 0x041F)); // xor 1
  return v;
}

__device__ __forceinline__ float artanhf_(float x) {
  x = fminf(fmaxf(x, -1.0f + 1e-7f), 1.0f - 1e-7f);
  return 0.5f * logf((1.0f + x) / (1.0f - x));
}

// ---------------------------------------------------------------------------
// Prep: hb = proj(expmap0(b)) for b1,b2 ; y2 = ||hb||^2. One wave.
// params layout: [0..63]=hb1, [64..127]=hb2, [128]=y2_1, [129]=y2_2
// ---------------------------------------------------------------------------
__global__ void prep_kernel(const float* __restrict__ b1,
                            const float* __restrict__ b2,
                            float* __restrict__ params) {
  const int lane = threadIdx.x & 31;
  const float* bs[2] = {b1, b2};
#pragma unroll
  for (int j = 0; j < 2; ++j) {
    float v0 = bs[j][lane];
    float v1 = bs[j][lane + 32];
    float n  = fmaxf(sqrtf(wsum(v0 * v0 + v1 * v1)), EPSV);
    float tn = fminf(tanhf(n), MAXN);
    float f  = tn / n;
    params[j * 64 + lane]      = f * v0;
    params[j * 64 + lane + 32] = f * v1;
    if (lane == 0) params[128 + j] = tn * tn;
  }
}

// ---------------------------------------------------------------------------
// HypLinear fused kernel: 16 rows/block, 4 waves, each wave does a 16x16
// output tile via V_WMMA_F32_16X16X4_F32 over K. Epilogue does
// mobius_matvec scaling + mobius_add(hb) + proj + logmap0, rowwise.
// FIRST: input is raw x [N,256]; expmap0+proj folded in as per-row scale f
// applied to mx (since (f*x)@W.T = f*(x@W.T)). Else input is h1 [N,64] and
// per-row norm comes from xnbuf.
// ---------------------------------------------------------------------------
template <bool FIRST, int K>
__global__ void __launch_bounds__(128) hyp_linear_kernel(
    const float* __restrict__ X, const float* __restrict__ W,
    const float* __restrict__ params, const float* __restrict__ xnbuf,
    float* __restrict__ xtout, int N, int hboff, int y2off) {
  constexpr int LDP = K + 4;        // pad: 4-bank stagger across rows
  __shared__ float ldsX[16 * LDP];
  __shared__ float ldsMX[16 * 68];
  __shared__ float ldsF[16];
  __shared__ float ldsXN[16];

  const int tid  = threadIdx.x;
  const int wave = tid >> 5;
  const int lane = tid & 31;
  const int r0   = blockIdx.x * 16;

  // cooperative load of 16 x K input tile (float4 per thread per iter)
#pragma unroll
  for (int i = 0; i < (16 * K) / (128 * 4); ++i) {
    int e = (tid + i * 128) * 4;
    int row = e / K, col = e % K;
    int rr = min(r0 + row, N - 1);
    const float4 v = *(const float4*)(X + (size_t)rr * K + col);
    float* p = &ldsX[row * LDP + col];
    p[0] = v.x; p[1] = v.y; p[2] = v.z; p[3] = v.w;
  }
  __syncthreads();

  if constexpr (FIRST) {
    if (tid < 16) {               // per-row: f = min(tanh n, MAXN)/n
      float s = 0.f;
      const float4* rp = (const float4*)&ldsX[tid * LDP];
#pragma unroll 4
      for (int k = 0; k < K / 4; ++k) {
        float4 u = rp[k];
        s += u.x * u.x + u.y * u.y + u.z * u.z + u.w * u.w;
      }
      float n  = fmaxf(sqrtf(s), EPSV);
      float cl = fminf(tanhf(n), MAXN);
      ldsF[tid]  = cl / n;
      ldsXN[tid] = fmaxf(cl, EPSV);
    }
    __syncthreads();
  }

  // ---- GEMM: mx[16][64], wave owns cols n0..n0+15 ----
  const int mrow  = lane & 15;          // A row / B col
  const int khalf = (lane >> 4) * 2;    // upper lanes hold K+2,K+3
  const int n0    = wave * 16;
  v8f acc = {0, 0, 0, 0, 0, 0, 0, 0};
  const float* wrow = W + (size_t)(n0 + mrow) * K + khalf;
#pragma unroll 4
  for (int kk = 0; kk < K; kk += 4) {
    v2f a, b;
    a[0] = ldsX[mrow * LDP + kk + khalf];
    a[1] = ldsX[mrow * LDP + kk + khalf + 1];
    b[0] = wrow[kk];
    b[1] = wrow[kk + 1];
    acc = __builtin_amdgcn_wmma_f32_16x16x4_f32(
        false, a, false, b, (short)0, acc, false, false);
  }
  const int rbase = (lane >> 4) * 8;    // C/D layout: VGPR v -> M = v (+8)
#pragma unroll
  for (int v = 0; v < 8; ++v)
    ldsMX[(v + rbase) * 68 + n0 + mrow] = acc[v];
  __syncthreads();

  // ---- rowwise epilogue: wave handles rows 4*wave .. 4*wave+3 ----
  const float y2  = params[y2off];
  const float hbl = params[hboff + lane];
  const float hbh = params[hboff + lane + 32];
#pragma unroll
  for (int i = 0; i < 4; ++i) {
    int r = wave * 4 + i;
    float m0 = ldsMX[r * 68 + lane];
    float m1 = ldsMX[r * 68 + lane + 32];
    float xnr;
    if constexpr (FIRST) {
      float f = ldsF[r]; m0 *= f; m1 *= f; xnr = ldsXN[r];
    } else {
      xnr = xnbuf[min(r0 + r, N - 1)];
    }
    float s2  = wsum(m0 * m0 + m1 * m1);
    float dd  = wsum(m0 * hbl + m1 * hbh);
    float mxn = fmaxf(sqrtf(s2), EPSV);
    // mobius_matvec scale + proj
    float t   = tanhf(mxn / xnr * artanhf_(xnr));
    float tcl = fminf(t, MAXN);
    float g   = tcl / mxn;                 // mv = g * mx ; ||mv|| = tcl
    // mobius_add(mv, hb)
    float x2  = tcl * tcl;
    float xy  = g * dd;
    float den = fmaxf(1.0f + 2.0f * xy + x2 * y2, EPSV);
    float cA  = (1.0f + 2.0f * xy + y2) / den * g;
    float cB  = (1.0f - x2) / den;
    float h0  = cA * m0 + cB * hbl;
    float h1  = cA * m1 + cB * hbh;
    // proj + logmap0
    float nh  = fmaxf(sqrtf(wsum(h0 * h0 + h1 * h1)), EPSV);
    float sp  = nh > MAXN ? (MAXN / nh) : 1.0f;
    float nph = fminf(nh, MAXN);
    float sl  = artanhf_(nph) / nph * sp;
    if (r0 + r < N) {
      xtout[(size_t)(r0 + r) * 64 + lane]      = sl * h0;
      xtout[(size_t)(r0 + r) * 64 + lane + 32] = sl * h1;
    }
  }
}

// ---------------------------------------------------------------------------
__global__ void zero_kernel(float4* __restrict__ p, long n4) {
  long i = (long)blockIdx.x * blockDim.x + threadIdx.x;
  if (i < n4) p[i] = make_float4(0.f, 0.f, 0.f, 0.f);
}

// one wave per edge; lane covers cols {2l, 2l+1}.
// Explicit GLOBAL_ATOMIC_ADD_F32 (no-return -> STOREcnt; agg fits in the
// 192MB L2 so these are L2-resident atomics). S_ENDPGM drains STOREcnt.
__global__ void __launch_bounds__(256) edge_kernel(
    const int* __restrict__ src, const int* __restrict__ dst,
    const float* __restrict__ w, const float* __restrict__ xt,
    float* __restrict__ agg, int E) {
  int e = blockIdx.x * 8 + (threadIdx.x >> 5);
  if (e >= E) return;
  int lane = threadIdx.x & 31;
  int s = src[e], d = dst[e];
  float we = w[e];
  const float2 v = *(const float2*)(xt + (size_t)s * 64 + lane * 2);
  float* ap = agg + (size_t)d * 64 + lane * 2;
  unsigned long long a64 = (unsigned long long)ap;
  float x0 = we * v.x;
  float x1 = we * v.y;
  asm volatile("global_atomic_add_f32 %0, %1, off\n\t"
               "global_atomic_add_f32 %0, %2, off offset:4"
               :: "v"(a64), "v"(x0), "v"(x1)
               : "memory");
}

// h = proj(expmap0(agg)); u = relu(logmap0(h)); out = proj(expmap0(u))
__global__ void __launch_bounds__(256) node_kernel(
    const float* __restrict__ agg, float* __restrict__ hout,
    float* __restrict__ xnout, int N) {
  int r = blockIdx.x * 8 + (threadIdx.x >> 5);
  if (r >= N) return;
  int lane = threadIdx.x & 31;
  float a0 = agg[(size_t)r * 64 + lane];
  float a1 = agg[(size_t)r * 64 + lane + 32];
  float n  = fmaxf(sqrtf(wsum(a0 * a0 + a1 * a1)), EPSV);
  float tn = fminf(tanhf(n), MAXN);
  float nh = fmaxf(tn, EPSV);
  float su = artanhf_(nh) / nh * (tn / n);   // logmap0(proj(expmap0(.)))
  float u0 = fmaxf(su * a0, 0.f);
  float u1 = fmaxf(su * a1, 0.f);
  float nr = fmaxf(sqrtf(wsum(u0 * u0 + u1 * u1)), EPSV);
  float tr = fminf(tanhf(nr), MAXN);
  float s2 = tr / nr;
  hout[(size_t)r * 64 + lane]      = s2 * u0;
  hout[(size_t)r * 64 + lane + 32] = s2 * u1;
  if (xnout != nullptr && lane == 0) xnout[r] = fmaxf(tr, EPSV);
}

// ---------------------------------------------------------------------------
extern "C" void kernel_launch(void* const* d_in, const int* in_sizes, int n_in,
                              void* d_out, int out_size, void* d_ws, size_t ws_size,
                              hipStream_t stream) {
  const float* x   = (const float*)d_in[0];
  const int*   src = (const int*)d_in[1];
  const int*   dst = (const int*)d_in[2];
  const float* ew  = (const float*)d_in[3];
  const float* W1  = (const float*)d_in[4];
  const float* b1  = (const float*)d_in[5];
  const float* W2  = (const float*)d_in[6];
  const float* b2  = (const float*)d_in[7];
  float* out = (float*)d_out;

  const int F = 256, D = 64;
  const int N = in_sizes[0] / F;
  const int E = in_sizes[1];

  float* ws     = (float*)d_ws;
  float* params = ws;                       // 256 floats
  float* xt     = ws + 256;                 // [N,64]
  float* agg    = xt + (size_t)N * D;       // [N,64]
  float* h1     = agg + (size_t)N * D;      // [N,64]
  float* xn     = h1 + (size_t)N * D;       // [N]

  const int gemmBlocks = (N + 15) / 16;
  const long n4 = (long)N * D / 4;
  const int zeroBlocks = (int)((n4 + 255) / 256);
  const int edgeBlocks = (E + 7) / 8;
  const int nodeBlocks = (N + 7) / 8;

  prep_kernel<<<1, 32, 0, stream>>>(b1, b2, params);

  // Layer 1
  hyp_linear_kernel<true, 256><<<gemmBlocks, 128, 0, stream>>>(
      x, W1, params, nullptr, xt, N, 0, 128);
  zero_kernel<<<zeroBlocks, 256, 0, stream>>>((float4*)agg, n4);
  edge_kernel<<<edgeBlocks, 256, 0, stream>>>(src, dst, ew, xt, agg, E);
  node_kernel<<<nodeBlocks, 256, 0, stream>>>(agg, h1, xn, N);

  // Layer 2
  hyp_linear_kernel<false, 64><<<gemmBlocks, 128, 0, stream>>>(
      h1, W2, params, xn, xt, N, 64, 129);
  zero_kernel<<<zeroBlocks, 256, 0, stream>>>((float4*)agg, n4);
  edge_kernel<<<edgeBlocks, 256, 0, stream>>>(src, dst, ew, xt, agg, E);
  node_kernel<<<nodeBlocks, 256, 0, stream>>>(agg, out, nullptr, N);
}